// SelectiveSSMBlock_953482739828
// MI455X (gfx1250) — compile-verified
//
#include <hip/hip_runtime.h>
#include <hip/hip_bf16.h>
#include <cmath>

// Selective SSM block for MI455X (gfx1250, wave32).
// - fp32 WMMA (V_WMMA_F32_16X16X4_F32): block is HBM/scan bound (~200MB @
//   23.3TB/s >> ~53 GFLOP of GEMM), so fp32 matrix path is free and exact.
// - B k-panels staged in LDS via gfx1250 async copies
//   (global_load_async_to_lds_b128 / s_wait_asynccnt), double-buffered:
//   removes the 8x duplicated B stream across the 8 waves of a block.

typedef __attribute__((ext_vector_type(2))) float v2f;
typedef __attribute__((ext_vector_type(8))) float v8f;

#define D_MODEL  1024
#define D_INNER  2048
#define D_STATE  16
#define DT_RANK  64
#define XP_COLS  96      // DT_RANK + 2*D_STATE
#define SEQ      2048
#define BATCH    2
#define MROWS    (BATCH * SEQ)   // 4096

#define KB    32    // K-panel depth staged in LDS
#define BROW  68    // padded LDS row stride (68 mod 64 = 4 -> conflict-free)

__device__ __forceinline__ float silu_f(float v) {
    return v / (1.0f + __expf(-v));
}
__device__ __forceinline__ float softplus_f(float v) {
    return fmaxf(v, 0.0f) + log1pf(__expf(-fabsf(v)));   // overflow-safe
}

// ---------------------------------------------------------------------------
// gfx1250 async global->LDS copy (ASYNCcnt). Inline asm: portable across
// ROCm-7.2 / amdgpu-toolchain builtin arity differences.
// vdst = LDS byte address (low 32 bits of generic pointer), vaddr = 64-bit
// global address.
// ---------------------------------------------------------------------------
__device__ __forceinline__ void async_copy_b128(uint32_t lds_addr,
                                                const float* gaddr) {
    asm volatile("global_load_async_to_lds_b128 %0, %1, off"
                 :: "v"(lds_addr), "v"(gaddr)
                 : "memory");
}
#define WAIT_ASYNCCNT(n) asm volatile("s_wait_asynccnt " #n ::: "memory")

// Cooperatively issue one KB x 64 B-panel (256 threads, 2 x b128 each).
template <int LDB>
__device__ __forceinline__ void issue_b_panel(const float* __restrict__ B,
                                              int k0, int col_base,
                                              float* dst) {
#pragma unroll
    for (int r = 0; r < 2; ++r) {
        const int id  = threadIdx.x + r * 256;    // 0..511
        const int row = id >> 4;                  // 0..31
        const int c4  = (id & 15) << 2;           // 0,4,...,60
        async_copy_b128((uint32_t)(size_t)(dst + row * BROW + c4),
                        B + (size_t)(k0 + row) * LDB + col_base + c4);
    }
}

// ---------------------------------------------------------------------------
// Block-level staged GEMM: 8 waves, block tile 128 x 64, wave tile 16 x 64.
// A streamed from global (contiguous float2 per lane), B consumed from the
// double-buffered LDS panel. WMMA fragment layouts per CDNA5 ISA 7.12.2:
//   A 16x4 f32: lane m=lane&15, K-pair k + 2*(lane>>4)  (contiguous pair)
//   B 4x16 f32: VGPR0 row K=k / K=k+2, VGPR1 row K=k+1 / K=k+3, N=lane&15
//   C/D 16x16:  VGPR v holds M=v (lanes 0-15) / M=v+8 (lanes 16-31)
// ---------------------------------------------------------------------------
template <int K, int LDA, int LDB>
__device__ __forceinline__ void block_gemm_staged(const float* __restrict__ A,
                                                  const float* __restrict__ B,
                                                  int row_base, int col_base,
                                                  v8f acc[4],
                                                  float bpan[2][KB * BROW]) {
    const int lane  = threadIdx.x & 31;
    const int m     = lane & 15;
    const int khalf = (lane >> 4) << 1;           // 0 or 2
    const float* arow = A + (size_t)(row_base + m) * LDA;
    constexpr int NP = K / KB;

    issue_b_panel<LDB>(B, 0, col_base, bpan[0]);
    for (int kp = 0; kp < NP; ++kp) {
        if (kp + 1 < NP) {
            issue_b_panel<LDB>(B, (kp + 1) * KB, col_base, bpan[(kp + 1) & 1]);
            WAIT_ASYNCCNT(0x2);   // own prior-panel copies drained (in-order)
        } else {
            WAIT_ASYNCCNT(0x0);
        }
        __syncthreads();          // whole panel visible to all waves

        const float* bb = bpan[kp & 1];
        const int k0 = kp * KB;
#pragma unroll
        for (int kl = 0; kl < KB; kl += 4) {
            const int kk = k0 + kl + khalf;
            v2f a;
            a.x = arow[kk];
            a.y = arow[kk + 1];
            const float* bl = bb + (kl + khalf) * BROW + m;
#pragma unroll
            for (int j = 0; j < 4; ++j) {
                v2f bf;
                bf.x = bl[j * 16];
                bf.y = bl[j * 16 + BROW];
                acc[j] = __builtin_amdgcn_wmma_f32_16x16x4_f32(
                    false, a, false, bf, (short)0, acc[j], false, false);
            }
        }
        __syncthreads();          // done reading before buffer reuse
    }
}

// ---------------------------------------------------------------------------
// K1: xz = x @ Win + bin ; split + SiLU.  Grid (32, 64), block 256.
// ---------------------------------------------------------------------------
__global__ __launch_bounds__(256) void gemm_xz_kernel(
    const float* __restrict__ x, const float* __restrict__ Win,
    const float* __restrict__ bin, float* __restrict__ xs,
    float* __restrict__ zs) {
    __shared__ float bpan[2][KB * BROW];
    const int lane = threadIdx.x & 31;
    const int wave = threadIdx.x >> 5;
    const int row_base = blockIdx.x * 128 + wave * 16;
    const int col_base = blockIdx.y * 64;
    v8f acc[4] = {};
    block_gemm_staged<D_MODEL, D_MODEL, 2 * D_INNER>(x, Win, row_base, col_base,
                                                     acc, bpan);
    const int mrow = row_base + ((lane >> 4) << 3);
    const int ncol = lane & 15;
#pragma unroll
    for (int v = 0; v < 8; ++v) {
        const int m = mrow + v;
#pragma unroll
        for (int j = 0; j < 4; ++j) {
            const int n = col_base + j * 16 + ncol;
            const float s = silu_f(acc[j][v] + bin[n]);
            if (n < D_INNER)
                xs[(size_t)m * D_INNER + n] = s;
            else
                zs[(size_t)m * D_INNER + (n - D_INNER)] = s;
        }
    }
}

// ---------------------------------------------------------------------------
// K2: xp = xs @ Wx + bx (4096 x 2048 x 96). N tiny -> simple global path.
// ---------------------------------------------------------------------------
__global__ __launch_bounds__(256) void gemm_xp_kernel(
    const float* __restrict__ xs, const float* __restrict__ Wx,
    const float* __restrict__ bx, float* __restrict__ xp) {
    const int lane = threadIdx.x & 31;
    const int wave = threadIdx.x >> 5;
    const int row_base = blockIdx.x * 128 + wave * 16;
    const int col_base = blockIdx.y * 16;
    const int m = lane & 15;
    const int khalf = (lane >> 4) << 1;
    const float* arow = xs + (size_t)(row_base + m) * D_INNER;
    v8f acc = {};
    for (int k = 0; k < D_INNER; k += 4) {
        const int kk = k + khalf;
        v2f a;
        a.x = arow[kk];
        a.y = arow[kk + 1];
        const float* bp = Wx + (size_t)kk * XP_COLS + col_base + m;
        v2f bf;
        bf.x = bp[0];
        bf.y = bp[XP_COLS];
        acc = __builtin_amdgcn_wmma_f32_16x16x4_f32(false, a, false, bf,
                                                    (short)0, acc, false, false);
    }
    const int mrow = row_base + ((lane >> 4) << 3);
    const int n = col_base + m;
#pragma unroll
    for (int v = 0; v < 8; ++v)
        xp[(size_t)(mrow + v) * XP_COLS + n] = acc[v] + bx[n];
}

// ---------------------------------------------------------------------------
// K3: delta = softplus(dp @ Wdt + bdt), dp = xp[:, :64] (lda = 96).
// ---------------------------------------------------------------------------
__global__ __launch_bounds__(256) void gemm_dt_kernel(
    const float* __restrict__ xp, const float* __restrict__ Wdt,
    const float* __restrict__ bdt, float* __restrict__ delta) {
    __shared__ float bpan[2][KB * BROW];
    const int lane = threadIdx.x & 31;
    const int wave = threadIdx.x >> 5;
    const int row_base = blockIdx.x * 128 + wave * 16;
    const int col_base = blockIdx.y * 64;
    v8f acc[4] = {};
    block_gemm_staged<DT_RANK, XP_COLS, D_INNER>(xp, Wdt, row_base, col_base,
                                                 acc, bpan);
    const int mrow = row_base + ((lane >> 4) << 3);
    const int ncol = lane & 15;
#pragma unroll
    for (int v = 0; v < 8; ++v) {
        const int m = mrow + v;
#pragma unroll
        for (int j = 0; j < 4; ++j) {
            const int n = col_base + j * 16 + ncol;
            delta[(size_t)m * D_INNER + n] = softplus_f(acc[j][v] + bdt[n]);
        }
    }
}

// ---------------------------------------------------------------------------
// K4: sequential selective scan. One 16-lane half-wave per (b,d) channel;
// lane = state index. Shuffle-reduce y over the 16 states (wave32-native).
// ---------------------------------------------------------------------------
__global__ __launch_bounds__(256) void ssm_scan_kernel(
    const float* __restrict__ xs, const float* __restrict__ zs,
    const float* __restrict__ xp, const float* __restrict__ delta,
    const float* __restrict__ A_log, const float* __restrict__ Dp,
    float* __restrict__ hidden, float* __restrict__ ypre) {
    const int tid   = threadIdx.x;
    const int n     = tid & 15;
    const int group = tid >> 4;
    const int c     = blockIdx.x * 16 + group;   // channel 0..4095
    const int b     = c >> 11;
    const int d     = c & (D_INNER - 1);

    const float Acoef = -__expf(A_log[d * D_STATE + n]);
    const float Dcoef = Dp[d];
    float h = 0.0f;

    for (int t = 0; t < SEQ; ++t) {
        const size_t row = (size_t)(b * SEQ + t);
        const float dt = delta[row * D_INNER + d];
        const float xt = xs[row * D_INNER + d];
        const float Bt = xp[row * XP_COLS + DT_RANK + n];
        const float Ct = xp[row * XP_COLS + DT_RANK + D_STATE + n];

        const float dA = __expf(dt * Acoef);
        h = dA * h + (dt * xt) * Bt;

        float p = h * Ct;
        p += __shfl_xor(p, 1, 32);
        p += __shfl_xor(p, 2, 32);
        p += __shfl_xor(p, 4, 32);
        p += __shfl_xor(p, 8, 32);

        if (n == 0) {
            hidden[row * D_INNER + d] = p;
            ypre[row * D_INNER + d] = (p + xt * Dcoef) * zs[row * D_INNER + d];
        }
    }
}

// ---------------------------------------------------------------------------
// K5: out = ypre @ Wout + bout (4096 x 2048 x 1024). Grid (32, 16).
// ---------------------------------------------------------------------------
__global__ __launch_bounds__(256) void gemm_out_kernel(
    const float* __restrict__ ypre, const float* __restrict__ Wout,
    const float* __restrict__ bout, float* __restrict__ out) {
    __shared__ float bpan[2][KB * BROW];
    const int lane = threadIdx.x & 31;
    const int wave = threadIdx.x >> 5;
    const int row_base = blockIdx.x * 128 + wave * 16;
    const int col_base = blockIdx.y * 64;
    v8f acc[4] = {};
    block_gemm_staged<D_INNER, D_INNER, D_MODEL>(ypre, Wout, row_base, col_base,
                                                 acc, bpan);
    const int mrow = row_base + ((lane >> 4) << 3);
    const int ncol = lane & 15;
#pragma unroll
    for (int v = 0; v < 8; ++v) {
        const int m = mrow + v;
#pragma unroll
        for (int j = 0; j < 4; ++j) {
            const int n = col_base + j * 16 + ncol;
            out[(size_t)m * D_MODEL + n] = acc[j][v] + bout[n];
        }
    }
}

extern "C" void kernel_launch(void* const* d_in, const int* in_sizes, int n_in,
                              void* d_out, int out_size, void* d_ws, size_t ws_size,
                              hipStream_t stream) {
    const float* x     = (const float*)d_in[0];
    const float* Win   = (const float*)d_in[1];
    const float* bin   = (const float*)d_in[2];
    const float* Wx    = (const float*)d_in[3];
    const float* bx    = (const float*)d_in[4];
    const float* Wdt   = (const float*)d_in[5];
    const float* bdt   = (const float*)d_in[6];
    const float* A_log = (const float*)d_in[7];
    const float* Dp    = (const float*)d_in[8];
    const float* Wout  = (const float*)d_in[9];
    const float* bout  = (const float*)d_in[10];

    float* out    = (float*)d_out;                          // (B,S,D_MODEL)
    float* hidden = out + (size_t)MROWS * D_MODEL;          // (B,S,D_INNER)

    float* ws    = (float*)d_ws;
    float* xs    = ws;                                      // 4096*2048
    float* zs    = xs    + (size_t)MROWS * D_INNER;         // 4096*2048
    float* xp    = zs    + (size_t)MROWS * D_INNER;         // 4096*96
    float* delta = xp    + (size_t)MROWS * XP_COLS;         // 4096*2048
    float* ypre  = delta + (size_t)MROWS * D_INNER;         // 4096*2048

    dim3 blk(256);
    gemm_xz_kernel<<<dim3(MROWS / 128, (2 * D_INNER) / 64), blk, 0, stream>>>(
        x, Win, bin, xs, zs);
    gemm_xp_kernel<<<dim3(MROWS / 128, XP_COLS / 16), blk, 0, stream>>>(
        xs, Wx, bx, xp);
    gemm_dt_kernel<<<dim3(MROWS / 128, D_INNER / 64), blk, 0, stream>>>(
        xp, Wdt, bdt, delta);
    ssm_scan_kernel<<<dim3((BATCH * D_INNER) / 16), blk, 0, stream>>>(
        xs, zs, xp, delta, A_log, Dp, hidden, ypre);
    gemm_out_kernel<<<dim3(MROWS / 128, D_MODEL / 64), blk, 0, stream>>>(
        ypre, Wout, bout, out);
}